// CrystalGraphConvNet_63273458204863
// MI455X (gfx1250) — compile-verified
//
#include <hip/hip_runtime.h>
#include <math.h>

typedef float v2f __attribute__((ext_vector_type(2)));
typedef float v8f __attribute__((ext_vector_type(8)));
typedef int   v4i __attribute__((ext_vector_type(4)));

constexpr int kN = 100000;   // atoms
constexpr int kM = 12;       // neighbors
constexpr int kC = 2000;     // crystals
constexpr int kORIG = 92;
constexpr int kNBR = 41;
constexpr int kF = 64;
constexpr int kH = 128;
constexpr float kEPS = 1e-5f;

#if __has_builtin(__builtin_amdgcn_global_load_async_to_lds_b128) && \
    __has_builtin(__builtin_amdgcn_s_wait_asynccnt)
#define USE_ASYNC_LDS 1
#else
#define USE_ASYNC_LDS 0
#endif

__device__ __forceinline__ float sigmoidf(float x) { return 1.f / (1.f + expf(-x)); }

// Stage W_edge [41,128] (padded to [44][132], zero rows 41..43) into LDS.
// Async direct-to-LDS path when the toolchain exposes the gfx1250 builtin.
__device__ __forceinline__ void stage_Wedge(float* Wl, const float* __restrict__ W, int tid) {
#if USE_ASYNC_LDS
  const int wv = tid >> 5, ln = tid & 31;
  for (int k = wv; k < kNBR; k += 8) {
    // each lane moves 16B: one row (128 floats) per wave-instruction
    __builtin_amdgcn_global_load_async_to_lds_b128(
        (__attribute__((address_space(1))) v4i*)(W + (size_t)(2 * kF + k) * 128 + ln * 4),
        (__attribute__((address_space(3))) v4i*)(Wl + k * 132 + ln * 4), 0, 0);
  }
  for (int idx = tid; idx < 3 * 132; idx += 256) Wl[kNBR * 132 + idx] = 0.f;
  __builtin_amdgcn_s_wait_asynccnt(0);
#else
  for (int idx = tid; idx < 44 * 128; idx += 256) {
    int k = idx >> 7, c = idx & 127;
    Wl[k * 132 + c] = (k < kNBR) ? W[(size_t)(2 * kF + k) * 128 + c] : 0.f;
  }
#endif
}

// ---------------------------------------------------------------------------
// Generic fp32 WMMA GEMM: D[rows,cols] = act(A[rows,K] @ B[K,cols](ldb) + bias)
// grid = (rows/16, ceil(cols/64)), block = 128 (4 waves, one 16-col tile each).
// K % 4 == 0, rows % 16 == 0. act: 0 = none, 1 = tanh, 2 = softplus
// ---------------------------------------------------------------------------
__global__ __launch_bounds__(128) void gemm16(
    const float* __restrict__ A, const float* __restrict__ B,
    const float* __restrict__ bias, float* __restrict__ D,
    int K, int ldb, int cols, int act)
{
  const int lane = threadIdx.x & 31;
  const int wave = threadIdx.x >> 5;
  const int ct   = blockIdx.y * 4 + wave;
  if (ct * 16 >= cols) return;
  const int n16  = lane & 15;
  const int kph  = (lane >> 4) * 2;          // K phase: lanes 0-15 -> K+0/1, 16-31 -> K+2/3
  const int rowBase = blockIdx.x * 16;
  const int colBase = ct * 16;
  const float* Arow = A + (size_t)(rowBase + n16) * K;

  v8f acc = {};
  for (int k0 = 0; k0 < K; k0 += 4) {
    v2f a, b;
    a.x = Arow[k0 + kph];
    a.y = Arow[k0 + kph + 1];
    b.x = B[(size_t)(k0 + kph)     * ldb + colBase + n16];
    b.y = B[(size_t)(k0 + kph + 1) * ldb + colBase + n16];
    acc = __builtin_amdgcn_wmma_f32_16x16x4_f32(false, a, false, b, (short)0, acc, false, false);
  }
  const int col = colBase + n16;
  const float bv = bias ? bias[col] : 0.f;
  const int rOff = rowBase + (lane >> 4) * 8;
#pragma unroll
  for (int i = 0; i < 8; ++i) {
    float v = acc[i] + bv;
    if (act == 1)      v = tanhf(v);
    else if (act == 2) v = (v > 20.f) ? v : log1pf(expf(v));
    D[(size_t)(rOff + i) * cols + col] = v;
  }
}

// ---------------------------------------------------------------------------
// Edge GEMM pass 1: g[e,c] = S[atom(e),c] + T[idx(e),c] + nbr_fea[e]@W_edge
// Accumulate per-column sum / sumsq for BatchNorm (over N*M rows).
// Block: 256 threads (8 waves), 32 atoms = 384 edges = 24 row tiles (3/wave).
// ---------------------------------------------------------------------------
__global__ __launch_bounds__(256) void edge_pass1(
    const float* __restrict__ nbr_fea, const int* __restrict__ nbr_idx,
    const float* __restrict__ S, const float* __restrict__ T,
    const float* __restrict__ W,     // conv_W layer base [169,128]; edge rows at +128
    float* __restrict__ gsum, float* __restrict__ gsq)
{
  __shared__ float Wl[44 * 132];     // padded [44][132], rows 41..43 zero
  __shared__ float ps[128], pq[128];
  const int tid = threadIdx.x;
  stage_Wedge(Wl, W, tid);
  if (tid < 128) { ps[tid] = 0.f; pq[tid] = 0.f; }
  __syncthreads();

  const int lane = tid & 31, wave = tid >> 5;
  const int n16 = lane & 15;
  const int kph = (lane >> 4) * 2;

  for (int t = 0; t < 3; ++t) {
    const long eBase = (long)blockIdx.x * 384 + (wave * 3 + t) * 16;
    const float* Ap = nbr_fea + (eBase + n16) * kNBR;
    if (t < 2) __builtin_prefetch(Ap + 16 * kNBR, 0, 0);   // next tile's row
    v2f af[11];
#pragma unroll
    for (int kk = 0; kk < 11; ++kk) {
      int k = kk * 4 + kph;
      af[kk].x = (k     < kNBR) ? Ap[k]     : 0.f;
      af[kk].y = (k + 1 < kNBR) ? Ap[k + 1] : 0.f;
    }
    int arow[8], jrow[8];
#pragma unroll
    for (int i = 0; i < 8; ++i) {
      long e2 = eBase + i + (lane >> 4) * 8;
      arow[i] = (int)(e2 / kM);
      jrow[i] = nbr_idx[e2];
    }
#pragma unroll
    for (int ct = 0; ct < 8; ++ct) {
      const int col = ct * 16 + n16;
      v8f acc = {};
#pragma unroll
      for (int kk = 0; kk < 11; ++kk) {
        v2f b;
        b.x = Wl[(kk * 4 + kph)     * 132 + col];
        b.y = Wl[(kk * 4 + kph + 1) * 132 + col];
        acc = __builtin_amdgcn_wmma_f32_16x16x4_f32(false, af[kk], false, b, (short)0, acc, false, false);
      }
      float s = 0.f, q = 0.f;
#pragma unroll
      for (int i = 0; i < 8; ++i) {
        float g = acc[i] + S[(size_t)arow[i] * 128 + col] + T[(size_t)jrow[i] * 128 + col];
        s += g; q += g * g;
      }
      atomicAdd(&ps[col], s);
      atomicAdd(&pq[col], q);
    }
  }
  __syncthreads();
  if (tid < 128) { atomicAdd(&gsum[tid], ps[tid]); atomicAdd(&gsq[tid], pq[tid]); }
}

// ---------------------------------------------------------------------------
// Edge GEMM pass 2: recompute g, apply BN (folded scale/shift), then
// sigmoid(g[:,:64]) * tanh(g[:,64:]) summed over M -> nbr_sumed [N,64].
// ---------------------------------------------------------------------------
__global__ __launch_bounds__(256) void edge_pass2(
    const float* __restrict__ nbr_fea, const int* __restrict__ nbr_idx,
    const float* __restrict__ S, const float* __restrict__ T,
    const float* __restrict__ W,
    const float* __restrict__ bscale, const float* __restrict__ bshift,
    float* __restrict__ ns)
{
  __shared__ float Wl[44 * 132];
  __shared__ float sc[128], sh[128];
  __shared__ float acb[32 * 64];     // per-atom accumulators
  const int tid = threadIdx.x;
  stage_Wedge(Wl, W, tid);
  if (tid < 128) { sc[tid] = bscale[tid]; sh[tid] = bshift[tid]; }
  for (int idx = tid; idx < 32 * 64; idx += 256) acb[idx] = 0.f;
  __syncthreads();

  const int lane = tid & 31, wave = tid >> 5;
  const int n16 = lane & 15;
  const int kph = (lane >> 4) * 2;
  const int aLocalBase = blockIdx.x * 32;

  for (int t = 0; t < 3; ++t) {
    const long eBase = (long)blockIdx.x * 384 + (wave * 3 + t) * 16;
    const float* Ap = nbr_fea + (eBase + n16) * kNBR;
    if (t < 2) __builtin_prefetch(Ap + 16 * kNBR, 0, 0);
    v2f af[11];
#pragma unroll
    for (int kk = 0; kk < 11; ++kk) {
      int k = kk * 4 + kph;
      af[kk].x = (k     < kNBR) ? Ap[k]     : 0.f;
      af[kk].y = (k + 1 < kNBR) ? Ap[k + 1] : 0.f;
    }
    int arow[8], jrow[8];
#pragma unroll
    for (int i = 0; i < 8; ++i) {
      long e2 = eBase + i + (lane >> 4) * 8;
      arow[i] = (int)(e2 / kM);
      jrow[i] = nbr_idx[e2];
    }
    float fb[4][8];                  // filter half, kept until matching core tile
#pragma unroll
    for (int ct = 0; ct < 8; ++ct) {
      const int col = ct * 16 + n16;
      v8f acc = {};
#pragma unroll
      for (int kk = 0; kk < 11; ++kk) {
        v2f b;
        b.x = Wl[(kk * 4 + kph)     * 132 + col];
        b.y = Wl[(kk * 4 + kph + 1) * 132 + col];
        acc = __builtin_amdgcn_wmma_f32_16x16x4_f32(false, af[kk], false, b, (short)0, acc, false, false);
      }
#pragma unroll
      for (int i = 0; i < 8; ++i) {
        float g  = acc[i] + S[(size_t)arow[i] * 128 + col] + T[(size_t)jrow[i] * 128 + col];
        float bn = g * sc[col] + sh[col];
        if (ct < 4) {
          fb[ct][i] = sigmoidf(bn);
        } else {
          float contrib = fb[ct - 4][i] * tanhf(bn);
          atomicAdd(&acb[(arow[i] - aLocalBase) * 64 + (col - 64)], contrib);
        }
      }
    }
  }
  __syncthreads();
  for (int idx = tid; idx < 32 * 64; idx += 256)
    ns[((size_t)aLocalBase + (idx >> 6)) * 64 + (idx & 63)] = acb[idx];
}

// ---------------------------------------------------------------------------
// Small helper kernels
// ---------------------------------------------------------------------------
__global__ void zerok(float* p, int n) {
  int i = blockIdx.x * blockDim.x + threadIdx.x;
  if (i < n) p[i] = 0.f;
}

// BN finalize: fold mean/var/gamma/beta into per-column scale/shift
__global__ void bn_fin(const float* s, const float* q, const float* g, const float* b,
                       float* scale, float* shift, float invcnt, int n) {
  int i = threadIdx.x;
  if (i < n) {
    float m  = s[i] * invcnt;
    float v  = q[i] * invcnt - m * m;
    float sc = g[i] * rsqrtf(v + kEPS);
    scale[i] = sc;
    shift[i] = b[i] - m * sc;
  }
}

// Per-column sum/sumsq of [N,64] matrix. grid=200 blocks x 64 threads, 500 rows each.
__global__ void colstats64(const float* __restrict__ x, float* s, float* q) {
  int col = threadIdx.x;
  long r0 = (long)blockIdx.x * 500;
  float ls = 0.f, lq = 0.f;
  for (int r = 0; r < 500; ++r) {
    float v = x[(r0 + r) * 64 + col];
    ls += v; lq += v * v;
  }
  atomicAdd(&s[col], ls);
  atomicAdd(&q[col], lq);
}

__global__ void update_x(float* x, const float* ns, const float* sc, const float* sh, int n) {
  int i = blockIdx.x * blockDim.x + threadIdx.x;
  if (i < n) {
    int c = i & 63;
    x[i] = tanhf(x[i] + ns[i] * sc[c] + sh[c]);
  }
}

__global__ void segsum(const float* __restrict__ x, const int* __restrict__ cid,
                       float* cs, float* cnt, int n64) {
  int i = blockIdx.x * blockDim.x + threadIdx.x;
  if (i < n64) {
    int n = i >> 6, c = i & 63;
    int g = cid[n];
    atomicAdd(&cs[(size_t)g * 64 + c], x[i]);
    if (c == 0) atomicAdd(&cnt[g], 1.f);
  }
}

__global__ void meandiv(const float* cs, const float* cnt, float* out, int n) {
  int i = blockIdx.x * blockDim.x + threadIdx.x;
  if (i < n) out[i] = cs[i] / fmaxf(cnt[i >> 6], 1.f);
}

__global__ void attn(const float* __restrict__ x, const float* __restrict__ cm2,
                     const int* __restrict__ cid, float* a, int n) {
  int i = blockIdx.x * blockDim.x + threadIdx.x;
  if (i < n) {
    const float* xr = x + (size_t)i * 64;
    const float* cr = cm2 + (size_t)cid[i] * 64;
    float s = 0.f;
#pragma unroll
    for (int k = 0; k < 64; ++k) s += xr[k] * cr[k];
    a[i] = sigmoidf(s);
  }
}

__global__ void wsum(const float* __restrict__ x, const float* __restrict__ a,
                     const int* __restrict__ cid, float* cs, int n64) {
  int i = blockIdx.x * blockDim.x + threadIdx.x;
  if (i < n64) {
    int n = i >> 6, c = i & 63;
    atomicAdd(&cs[(size_t)cid[n] * 64 + c], a[n] * x[i]);
  }
}

__global__ void outk(const float* __restrict__ h, const float* __restrict__ w,
                     const float* __restrict__ b, float* out, int c) {
  int i = blockIdx.x * blockDim.x + threadIdx.x;
  if (i < c) {
    float s = b[0];
#pragma unroll
    for (int k = 0; k < 128; ++k) s += h[(size_t)i * 128 + k] * w[k];
    out[i] = s;
  }
}

// ---------------------------------------------------------------------------
extern "C" void kernel_launch(void* const* d_in, const int* in_sizes, int n_in,
                              void* d_out, int out_size, void* d_ws, size_t ws_size,
                              hipStream_t stream) {
  const float* atom_fea = (const float*)d_in[0];
  const float* nbr_fea  = (const float*)d_in[1];
  const int*   nbr_idx  = (const int*)d_in[2];
  const int*   cids     = (const int*)d_in[3];
  const float* emb_W    = (const float*)d_in[4];
  const float* emb_b    = (const float*)d_in[5];
  const float* conv_W   = (const float*)d_in[6];
  const float* conv_b   = (const float*)d_in[7];
  const float* bn1_g    = (const float*)d_in[8];
  const float* bn1_b    = (const float*)d_in[9];
  const float* bn2_g    = (const float*)d_in[10];
  const float* bn2_b    = (const float*)d_in[11];
  const float* W3_W     = (const float*)d_in[12];
  const float* W3_b     = (const float*)d_in[13];
  const float* fc_W     = (const float*)d_in[14];
  const float* fc_b     = (const float*)d_in[15];
  const float* out_W    = (const float*)d_in[16];
  const float* out_b    = (const float*)d_in[17];

  float* out = (float*)d_out;           // [C]
  float* x   = out + kC;                // [N,64] — second reference output, computed in place

  // workspace layout
  float* S    = (float*)d_ws;                        // [N,128]
  float* T    = S + (size_t)kN * 128;                // [N,128]
  float* NS   = T + (size_t)kN * 128;                // [N,64]
  float* stat = NS + (size_t)kN * 64;                // 1024: s1,q1,scale1,shift1,s2,q2,scale2,shift2
  float* csum = stat + 1024;                         // [C,64]
  float* cnt  = csum + (size_t)kC * 64;              // [C]
  float* cmean= cnt + kC;                            // [C,64]
  float* cm2  = cmean + (size_t)kC * 64;             // [C,64]
  float* aatt = cm2 + (size_t)kC * 64;               // [N]
  float* cf   = aatt + kN;                           // [C,64]
  float* hbuf = cf + (size_t)kC * 64;                // [C,128]

  // ---- embedding: x = atom_fea @ emb_W + emb_b ----
  gemm16<<<dim3(kN / 16, 1), 128, 0, stream>>>(atom_fea, emb_W, emb_b, x,
                                               kORIG, kF, kF, 0);

  // ---- 3 conv layers ----
  for (int l = 0; l < 3; ++l) {
    const float* Wl = conv_W + (size_t)l * (2 * kF + kNBR) * 128;
    const float* bl = conv_b + (size_t)l * 128;
    // S = x @ W[:64] + b ; T = x @ W[64:128]
    gemm16<<<dim3(kN / 16, 2), 128, 0, stream>>>(x, Wl,                  bl,      S, kF, 128, 128, 0);
    gemm16<<<dim3(kN / 16, 2), 128, 0, stream>>>(x, Wl + (size_t)kF*128, nullptr, T, kF, 128, 128, 0);
    // BN1 stats (two-pass, no materialized [N*M,128])
    zerok<<<1, 256, 0, stream>>>(stat, 256);
    edge_pass1<<<kN * kM / 384, 256, 0, stream>>>(nbr_fea, nbr_idx, S, T, Wl,
                                                  stat, stat + 128);
    bn_fin<<<1, 128, 0, stream>>>(stat, stat + 128, bn1_g + l * 128, bn1_b + l * 128,
                                  stat + 256, stat + 384, 1.f / (float)(kN * kM), 128);
    edge_pass2<<<kN * kM / 384, 256, 0, stream>>>(nbr_fea, nbr_idx, S, T, Wl,
                                                  stat + 256, stat + 384, NS);
    // BN2 over nbr_sumed, then x = tanh(x + bn2(ns))
    zerok<<<1, 128, 0, stream>>>(stat + 512, 128);
    colstats64<<<200, 64, 0, stream>>>(NS, stat + 512, stat + 576);
    bn_fin<<<1, 64, 0, stream>>>(stat + 512, stat + 576, bn2_g + l * 64, bn2_b + l * 64,
                                 stat + 640, stat + 704, 1.f / (float)kN, 64);
    update_x<<<(kN * 64 + 255) / 256, 256, 0, stream>>>(x, NS, stat + 640, stat + 704, kN * 64);
  }

  // ---- pooling / attention head ----
  zerok<<<(kC * 64 + kC + 255) / 256, 256, 0, stream>>>(csum, kC * 64 + kC);
  segsum<<<(kN * 64 + 255) / 256, 256, 0, stream>>>(x, cids, csum, cnt, kN * 64);
  meandiv<<<(kC * 64 + 255) / 256, 256, 0, stream>>>(csum, cnt, cmean, kC * 64);
  // cm2 = tanh(crys_mean @ W3 + b3)  [C,64]
  gemm16<<<dim3(kC / 16, 1), 128, 0, stream>>>(cmean, W3_W, W3_b, cm2, kF, kF, kF, 1);
  attn<<<(kN + 255) / 256, 256, 0, stream>>>(x, cm2, cids, aatt, kN);
  zerok<<<(kC * 64 + 255) / 256, 256, 0, stream>>>(csum, kC * 64);
  wsum<<<(kN * 64 + 255) / 256, 256, 0, stream>>>(x, aatt, cids, csum, kN * 64);
  meandiv<<<(kC * 64 + 255) / 256, 256, 0, stream>>>(csum, cnt, cf, kC * 64);
  // hbuf = softplus(cf @ fc_W + fc_b)  [C,128]
  gemm16<<<dim3(kC / 16, 2), 128, 0, stream>>>(cf, fc_W, fc_b, hbuf, kF, kH, kH, 2);
  outk<<<(kC + 255) / 256, 256, 0, stream>>>(hbuf, out_W, out_b, out, kC);
}